// Calc_Attention_51032801411467
// MI455X (gfx1250) — compile-verified
//
#include <hip/hip_runtime.h>

// ---------------------------------------------------------------------------
// Fused attention block for MI455X (gfx1250, wave32, WMMA + TDM).
//   b=2, n=2048, dim=1024, heads=32, dim_head=32.  ~70 GFLOP fp32.
// All GEMMs on the matrix pipe via V_WMMA_F32_16X16X4_F32 (full f32 precision,
// matches the fp32 reference). Flash-style attention (no 1GB score tensor).
// Intermediates live in d_ws, L2-resident (192MB L2).
//
// Data movement:
//  * B/V operand tiles are stored in LDS as interleaved k-pairs
//    sB[k/2][2*n + (k&1)] so each WMMA B fragment {B[k][n],B[k+1][n]} is one
//    aligned ds_load_b64 into a natural even VGPR pair (verified last round:
//    inner loop is pure ds_load_2addr_b64 + v_wmma, no movs).
//  * A/K tiles are plain strided row copies with a +4-dword row pad -> staged
//    by the Tensor Data Mover (tensor_load_to_lds, pad_interval=32 dwords,
//    pad_amount=4 dwords reproduces the 36-float padded stride in HW),
//    completion via s_wait_tensorcnt, double-buffered in the GEMM so the DMA
//    for chunk i+1 overlaps chunk i's WMMA work.
// ---------------------------------------------------------------------------

typedef float v2f __attribute__((ext_vector_type(2)));
typedef float v4f __attribute__((ext_vector_type(4)));
typedef float v8f __attribute__((ext_vector_type(8)));
typedef unsigned int u32x4 __attribute__((ext_vector_type(4)));
typedef int i32x4 __attribute__((ext_vector_type(4)));
typedef int i32x8 __attribute__((ext_vector_type(8)));

// V_WMMA_F32_16X16X4_F32: D[16x16] = A[16x4] * B[4x16] + C
// A layout: lane L holds A[L%16][2*(L/16) + {0,1}] per k-step
// B layout: lane L holds B[2*(L/16) + {0,1}][L%16]
// C layout: lane L, vgpr v holds C[v + 8*(L/16)][L%16]
#define WMMA_F32(a, b, c) \
  __builtin_amdgcn_wmma_f32_16x16x4_f32(false, (a), false, (b), (short)0, (c), false, false)

#define V8F_ZERO ((v8f){0.f,0.f,0.f,0.f,0.f,0.f,0.f,0.f})

#if defined(__has_builtin)
#if __has_builtin(__builtin_amdgcn_tensor_load_to_lds) && \
    __has_builtin(__builtin_amdgcn_s_wait_tensorcnt)
#define HAVE_TDM 1
#endif
#endif
#ifndef HAVE_TDM
#define HAVE_TDM 0
#endif

#if HAVE_TDM
// --- Tensor DMA descriptor builders (CDNA5 ISA ch.8, D# groups 0/1) --------
// group0: [1:0]=count=1 | [63:32]=lds byte addr | [120:64]=global addr |
//         [127:126]=type=2
__device__ __forceinline__ u32x4 tdm_group0(const void* gptr, void* lds_ptr) {
  unsigned long long ga = (unsigned long long)gptr;
  // low 32 bits of a flat LDS address are the LDS byte offset (ISA 10.2)
  unsigned lds = (unsigned)(unsigned long long)lds_ptr;
  u32x4 g;
  g[0] = 1u;
  g[1] = lds;
  g[2] = (unsigned)(ga & 0xffffffffu);
  g[3] = (unsigned)((ga >> 32) & 0x01ffffffu) | (2u << 30);
  return g;
}
// group1 for a 2D tile of 4-byte elements with LDS row padding:
//   w0: data_size=4B (2<<16) | pad_enable (1<<20) | pad_interval<<22 |
//       pad_amount<<25      (interval code 4 = 32 dwords, amount code 3 = 4)
//   tensor_dim0/1 set large (tiles never clip), dim0 stride in elements.
__device__ __forceinline__ i32x8 tdm_group1_2d(unsigned tile_d0, unsigned tile_d1,
                                               unsigned long long stride0_elems) {
  const unsigned td0 = 0x100000u, td1 = 0x100000u;
  i32x8 g;
  g[0] = (int)((2u << 16) | (1u << 20) | (4u << 22) | (3u << 25));
  g[1] = (int)((td0 & 0xffffu) << 16);                    // tensor_dim0[15:0]
  g[2] = (int)(((td0 >> 16) & 0xffffu) | ((td1 & 0xffffu) << 16));
  g[3] = (int)(((td1 >> 16) & 0xffffu) | ((tile_d0 & 0xffffu) << 16));
  g[4] = (int)(tile_d1 & 0xffffu);                        // tile_dim2 = 0
  g[5] = (int)(stride0_elems & 0xffffffffu);
  g[6] = (int)((stride0_elems >> 32) & 0xffffu);          // dim1 stride = 0
  g[7] = 0;
  return g;
}
__device__ __forceinline__ void tdm_load_2d(u32x4 g0, i32x8 g1) {
  i32x4 z4 = (i32x4){0, 0, 0, 0};
#if __clang_major__ >= 23
  i32x8 z8 = (i32x8){0, 0, 0, 0, 0, 0, 0, 0};
  __builtin_amdgcn_tensor_load_to_lds(g0, g1, z4, z4, z8, 0);
#else
  __builtin_amdgcn_tensor_load_to_lds(g0, g1, z4, z4, 0);
#endif
}
#endif  // HAVE_TDM

__device__ __forceinline__ float red_max16(float x) {
  #pragma unroll
  for (int m = 1; m < 16; m <<= 1) x = fmaxf(x, __shfl_xor(x, m, 32));
  return x;
}
__device__ __forceinline__ float red_sum16(float x) {
  #pragma unroll
  for (int m = 1; m < 16; m <<= 1) x += __shfl_xor(x, m, 32);
  return x;
}
__device__ __forceinline__ float red_sum32(float x) {
  #pragma unroll
  for (int m = 1; m < 32; m <<= 1) x += __shfl_xor(x, m, 32);
  return x;
}

// ---------------------------------------------------------------------------
// Generic GEMM:  C[M,N] = A[M,K] @ B[K,N] (+ bias[N] if bias != null)
// 256 threads = 8 waves; tile 128x128; K-chunk 32. Wave w owns rows
// [w*16,w*16+16) x 128 cols; all waves share the B LDS block.
//   sA: [2][128][36] double-buffered, TDM-staged (DMA for chunk i+1 overlaps
//       chunk i compute); A frag = 1 ds_load_b64
//   sB: [16 pairs][288] interleaved k-pairs; B frag = 1 ds_load_b64
//       (288 mod 64 == 32 -> lane-halves use disjoint bank halves)
// ---------------------------------------------------------------------------
#define G_BM 128
#define G_BN 128
#define G_BK 32
#define G_LDA 36
#define G_LDB 288   // 2*128 + 32

__global__ __launch_bounds__(256) void gemm_wmma_kernel(
    const float* __restrict__ A, const float* __restrict__ B,
    const float* __restrict__ bias, float* __restrict__ C,
    int Mdim, int Ndim, int Kdim) {
  __shared__ float sA[2][G_BM * G_LDA];       // 36.9 KB
  __shared__ float sB[(G_BK / 2) * G_LDB];    // 18.4 KB

  const int tid   = threadIdx.x;
  const int wave  = tid >> 5;
  const int lane  = tid & 31;
  const int lhalf = lane >> 4;
  const int l16   = lane & 15;
  const int rBlk  = blockIdx.x * G_BM;
  const int cBlk  = blockIdx.y * G_BN;
  const int nChunks = Kdim / G_BK;

  v8f acc[8];
  #pragma unroll
  for (int j = 0; j < 8; ++j) acc[j] = V8F_ZERO;

#if HAVE_TDM
  const i32x8 g1a = tdm_group1_2d(G_BK, G_BM, (unsigned long long)Kdim);
  if (wave == 0) {
    tdm_load_2d(tdm_group0(&A[rBlk * Kdim], &sA[0][0]), g1a);
  }
#endif

  for (int c = 0; c < nChunks; ++c) {
    const int k0 = c * G_BK;
    __syncthreads();   // previous chunk's compute done reading sA/sB

#if !HAVE_TDM
    // Fallback: manual A staging (float4 rows)
    #pragma unroll
    for (int i = 0; i < 4; ++i) {
      int idx = i * 256 + tid;
      int r = idx >> 3, g = idx & 7;
      *(v4f*)&sA[c & 1][r * G_LDA + 4 * g] =
          *(const v4f*)&A[(rBlk + r) * Kdim + k0 + 4 * g];
    }
#endif
    // Stage B: 16 k-pairs x 128 cols, interleaved {k,k+1} per col
    #pragma unroll
    for (int i = 0; i < 2; ++i) {
      int idx = i * 256 + tid;
      int p = idx >> 5, g = idx & 31;
      const float* wp = &B[(k0 + 2 * p) * Ndim + cBlk + 4 * g];
      v4f w0 = *(const v4f*)wp;
      v4f w1 = *(const v4f*)(wp + Ndim);
      v4f lo = (v4f){w0.x, w1.x, w0.y, w1.y};
      v4f hi = (v4f){w0.z, w1.z, w0.w, w1.w};
      *(v4f*)&sB[p * G_LDB + 8 * g]     = lo;
      *(v4f*)&sB[p * G_LDB + 8 * g + 4] = hi;
    }
#if HAVE_TDM
    if (wave == 0) __builtin_amdgcn_s_wait_tensorcnt(0);
#endif
    __syncthreads();   // sA[c&1] (TDM) + sB stores visible to all waves

#if HAVE_TDM
    // Prefetch next A chunk into the other buffer; overlaps with compute.
    if (wave == 0 && (c + 1) < nChunks) {
      tdm_load_2d(tdm_group0(&A[rBlk * Kdim + k0 + G_BK], &sA[(c + 1) & 1][0]),
                  g1a);
    }
#endif

    const float* sAc = &sA[c & 1][0];
    #pragma unroll
    for (int s = 0; s < G_BK / 4; ++s) {
      const int kb = 4 * s + 2 * lhalf;     // even
      const int pp = 2 * s + lhalf;         // pair index
      v2f a = *(const v2f*)&sAc[(wave * 16 + l16) * G_LDA + kb];
      #pragma unroll
      for (int j = 0; j < 8; ++j) {
        v2f b = *(const v2f*)&sB[pp * G_LDB + 32 * j + 2 * l16];
        acc[j] = WMMA_F32(a, b, acc[j]);
      }
    }
  }

  const int row0 = rBlk + wave * 16 + 8 * lhalf;
  #pragma unroll
  for (int j = 0; j < 8; ++j) {
    const int col = cBlk + j * 16 + l16;
    const float bb = bias ? bias[col] : 0.f;
    #pragma unroll
    for (int v = 0; v < 8; ++v) {
      C[(row0 + v) * Ndim + col] = acc[j][v] + bb;
    }
  }
}

// ---------------------------------------------------------------------------
// Flash attention. One workgroup per (b*h, 128-query block).
// qkv layout: [b*2048 + n][ t*1024 + h*32 + d ], t in {q,k,v}.
// Each wave owns 16 query rows; loop over 32 key-blocks of 64 keys.
//   sK : [64][36] row-major, TDM-staged (DMA overlaps manual V staging)
//   sVp: [32 pairs][96] interleaved k-pairs (96 mod 64 == 32)
//   sP : per-wave [16][68] strip for C-layout -> A-layout round trip
// ---------------------------------------------------------------------------
#define A_KB  64
#define A_QB  128
#define A_LDK 36
#define A_LDV 96    // 2*32 + 32
#define A_LDP 68

__global__ __launch_bounds__(256) void attn_kernel(
    const float* __restrict__ QKV, float* __restrict__ ATT) {
  __shared__ float sK[A_KB * A_LDK];             // 9.2 KB
  __shared__ float sVp[(A_KB / 2) * A_LDV];      // 12.3 KB
  __shared__ float sP[8 * 16 * A_LDP];           // 34.8 KB

  const int tid   = threadIdx.x;
  const int wave  = tid >> 5;
  const int lane  = tid & 31;
  const int lhalf = lane >> 4;
  const int l16   = lane & 15;

  const int bh   = blockIdx.x;        // 0..63
  const int b    = bh >> 5;
  const int h    = bh & 31;
  const int qblk = blockIdx.y * A_QB;
  const float scale = 0.1767766952966369f;  // 32^-0.5

  // Preload this wave's 16 q rows in WMMA-A layout, pre-scaled (b64 loads).
  const int qrow = qblk + wave * 16 + l16;
  const float* qptr = QKV + (b * 2048 + qrow) * 3072 + h * 32;
  v2f qa[8];
  #pragma unroll
  for (int s = 0; s < 8; ++s) {
    v2f q2 = *(const v2f*)(qptr + 4 * s + 2 * lhalf);
    qa[s].x = q2.x * scale;
    qa[s].y = q2.y * scale;
  }

#if HAVE_TDM
  const i32x8 g1k = tdm_group1_2d(32, A_KB, 3072ull);
#endif

  v8f o0 = V8F_ZERO, o1 = V8F_ZERO;
  float rmax[8], rsum[8];
  #pragma unroll
  for (int v = 0; v < 8; ++v) { rmax[v] = -3.0e38f; rsum[v] = 0.f; }

  float* pw = sP + wave * 16 * A_LDP;

  for (int kb0 = 0; kb0 < 2048; kb0 += A_KB) {
    __syncthreads();
#if HAVE_TDM
    // Stage K [64x32] via Tensor Data Mover; overlaps the manual V staging.
    if (wave == 0) {
      const float* kp = QKV + (b * 2048 + kb0) * 3072 + h * 32 + 1024;
      tdm_load_2d(tdm_group0(kp, &sK[0]), g1k);
    }
#else
    #pragma unroll
    for (int i = 0; i < 2; ++i) {
      int idx = i * 256 + tid;
      int kr = idx >> 3, g = idx & 7;
      const float* kp = QKV + (b * 2048 + kb0 + kr) * 3072 + h * 32 + 1024 + 4 * g;
      *(v4f*)&sK[kr * A_LDK + 4 * g] = *(const v4f*)kp;
    }
#endif
    // Stage V as interleaved k-pairs: 32 pairs x 8 float4-groups = 256 tasks
    {
      int p = tid >> 3, g = tid & 7;
      const float* vp = QKV + (b * 2048 + kb0 + 2 * p) * 3072 + h * 32 + 2048 + 4 * g;
      v4f v0 = *(const v4f*)vp;
      v4f v1 = *(const v4f*)(vp + 3072);
      v4f lo = (v4f){v0.x, v1.x, v0.y, v1.y};
      v4f hi = (v4f){v0.z, v1.z, v0.w, v1.w};
      *(v4f*)&sVp[p * A_LDV + 8 * g]     = lo;
      *(v4f*)&sVp[p * A_LDV + 8 * g + 4] = hi;
    }
#if HAVE_TDM
    if (wave == 0) __builtin_amdgcn_s_wait_tensorcnt(0);
#endif
    __syncthreads();

    // ---- S = q @ K^T : 4 tiles of 16 keys, d=32 -> 8 wmma steps ----
    v8f st[4];
    #pragma unroll
    for (int j = 0; j < 4; ++j) st[j] = V8F_ZERO;
    #pragma unroll
    for (int s = 0; s < 8; ++s) {
      const int kd = 4 * s + 2 * lhalf;
      v2f a = qa[s];
      #pragma unroll
      for (int j = 0; j < 4; ++j) {
        v2f bb = *(const v2f*)&sK[(j * 16 + l16) * A_LDK + kd];
        st[j] = WMMA_F32(a, bb, st[j]);
      }
    }

    // ---- online softmax + rescale accumulators ----
    #pragma unroll
    for (int v = 0; v < 8; ++v) {
      float mx = fmaxf(fmaxf(st[0][v], st[1][v]), fmaxf(st[2][v], st[3][v]));
      mx = red_max16(mx);
      const float mnew  = fmaxf(rmax[v], mx);
      const float alpha = __expf(rmax[v] - mnew);
      rmax[v] = mnew;
      float ps = 0.f;
      #pragma unroll
      for (int j = 0; j < 4; ++j) {
        float p = __expf(st[j][v] - mnew);
        st[j][v] = p;
        ps += p;
      }
      ps = red_sum16(ps);
      rsum[v] = rsum[v] * alpha + ps;
      o0[v] *= alpha;
      o1[v] *= alpha;
    }

    // ---- C-layout -> A-layout via per-wave LDS strip ----
    #pragma unroll
    for (int j = 0; j < 4; ++j)
      #pragma unroll
      for (int v = 0; v < 8; ++v)
        pw[(v + 8 * lhalf) * A_LDP + j * 16 + l16] = st[j][v];
    // same-wave LDS ops are in-order; compiler orders store->load via deps

    // ---- O += P[16x64] @ V[64x32] : 16 wmma steps x 2 output tiles ----
    #pragma unroll
    for (int s = 0; s < 16; ++s) {
      const int kk = 4 * s + 2 * lhalf;
      const int pp = 2 * s + lhalf;
      v2f a  = *(const v2f*)&pw[l16 * A_LDP + kk];
      v2f b0 = *(const v2f*)&sVp[pp * A_LDV + 2 * l16];
      v2f b1 = *(const v2f*)&sVp[pp * A_LDV + 2 * (16 + l16)];
      o0 = WMMA_F32(a, b0, o0);
      o1 = WMMA_F32(a, b1, o1);
    }
  }

  // finalize: divide by row sums, store ATT[b*2048+row][h*32 + d]
  #pragma unroll
  for (int v = 0; v < 8; ++v) {
    const float inv = 1.f / rsum[v];
    const int row = qblk + wave * 16 + v + 8 * lhalf;
    float* op = ATT + (b * 2048 + row) * 1024 + h * 32;
    op[l16]      = o0[v] * inv;
    op[16 + l16] = o1[v] * inv;
  }
}

// ---------------------------------------------------------------------------
// LayerNorm over rows of Y[4096,1024]. One workgroup per row; each thread
// owns one float4. Memory-bound: ~34 MB total => ~1.5us at HBM rate.
// ---------------------------------------------------------------------------
__global__ __launch_bounds__(256) void ln_kernel(
    const float* __restrict__ Y, const float* __restrict__ gamma,
    const float* __restrict__ beta, float* __restrict__ OUT) {
  __shared__ float redS[8];
  __shared__ float redQ[8];
  const int tid  = threadIdx.x;
  const int wave = tid >> 5;
  const int lane = tid & 31;
  const int row  = blockIdx.x;

  v4f t = *(const v4f*)&Y[row * 1024 + 4 * tid];
  float s = t.x + t.y + t.z + t.w;
  float q = t.x * t.x + t.y * t.y + t.z * t.z + t.w * t.w;
  s = red_sum32(s);
  q = red_sum32(q);
  if (lane == 0) { redS[wave] = s; redQ[wave] = q; }
  __syncthreads();
  float S = 0.f, Q = 0.f;
  #pragma unroll
  for (int w = 0; w < 8; ++w) { S += redS[w]; Q += redQ[w]; }
  const float mean = S * (1.0f / 1024.0f);
  const float var  = Q * (1.0f / 1024.0f) - mean * mean;
  const float rstd = rsqrtf(var + 1e-6f);

  v4f g4 = *(const v4f*)&gamma[4 * tid];
  v4f b4 = *(const v4f*)&beta[4 * tid];
  v4f o;
  o.x = (t.x - mean) * rstd * g4.x + b4.x;
  o.y = (t.y - mean) * rstd * g4.y + b4.y;
  o.z = (t.z - mean) * rstd * g4.z + b4.z;
  o.w = (t.w - mean) * rstd * g4.w + b4.w;
  *(v4f*)&OUT[row * 1024 + 4 * tid] = o;
}

// ---------------------------------------------------------------------------
// Launch. Workspace: qkv [4096*3072] f32 (50.3MB) + att [4096*1024] f32
// (16.8MB) = 67.1MB. The projection output Y reuses the qkv region (qkv is
// dead after attn_kernel), keeping peak ws at 67.1MB.
// ---------------------------------------------------------------------------
extern "C" void kernel_launch(void* const* d_in, const int* in_sizes, int n_in,
                              void* d_out, int out_size, void* d_ws, size_t ws_size,
                              hipStream_t stream) {
  const float* x      = (const float*)d_in[0];
  const float* w_qkv  = (const float*)d_in[1];
  const float* w_out  = (const float*)d_in[2];
  const float* b_out  = (const float*)d_in[3];
  const float* lgamma = (const float*)d_in[4];
  const float* lbeta  = (const float*)d_in[5];

  float* qkv = (float*)d_ws;
  float* att = qkv + (size_t)4096 * 3072;
  float* y   = qkv;   // reuse (qkv dead after attention)

  gemm_wmma_kernel<<<dim3(4096 / G_BM, 3072 / G_BN), 256, 0, stream>>>(
      x, w_qkv, nullptr, qkv, 4096, 3072, 1024);
  attn_kernel<<<dim3(64, 2048 / A_QB), 256, 0, stream>>>(qkv, att);
  gemm_wmma_kernel<<<dim3(4096 / G_BM, 1024 / G_BN), 256, 0, stream>>>(
      att, w_out, b_out, y, 4096, 1024, 1024);
  ln_kernel<<<dim3(4096), 256, 0, stream>>>(y, lgamma, lbeta, (float*)d_out);
}